// CompletionNet_44693429682756
// MI455X (gfx1250) — compile-verified
//
#include <hip/hip_runtime.h>
#include <hip/hip_bf16.h>
#include <math.h>

#define KVOL 27
#define NPTS 1000000
#define NCH  16
#define TILES_PER_K (NPTS / 16)          // 62500, exact
#define BN_EPS 1e-5f

typedef float v2f __attribute__((ext_vector_type(2)));
typedef float v8f __attribute__((ext_vector_type(8)));

// ---------------- conv1: feats[N,1] x W1[K,1,16], gather/scatter ----------------
__global__ void conv1_scatter(const float* __restrict__ feats,
                              const float* __restrict__ W1,
                              const int* __restrict__ in_map,
                              const int* __restrict__ out_map,
                              float* __restrict__ out1, long total)
{
    long p = (long)blockIdx.x * blockDim.x + threadIdx.x;
    if (p >= total) return;
    int k = (int)(p / NPTS);
    float f = feats[in_map[p]];
    long orow = (long)out_map[p] * NCH;
    const float* w = W1 + (size_t)k * NCH;
#pragma unroll
    for (int c = 0; c < NCH; ++c)
        atomicAdd(&out1[orow + c], f * w[c]);
}

// ---------------- per-channel sum / sumsq over [N,16] ----------------
__global__ void reduce_stats(const float* __restrict__ x, float* __restrict__ stats)
{
    __shared__ float ssum[256], ssq[256];
    const int c = threadIdx.x & 15;
    const int g = threadIdx.x >> 4;      // row-group 0..15
    float s = 0.f, q = 0.f;
    for (long r = (long)blockIdx.x * 16 + g; r < NPTS; r += (long)gridDim.x * 16) {
        float v = x[r * NCH + c];
        s += v; q += v * v;
    }
    ssum[threadIdx.x] = s; ssq[threadIdx.x] = q;
    __syncthreads();
    if (g == 0) {
        float ts = 0.f, tq = 0.f;
#pragma unroll
        for (int j = 0; j < 16; ++j) { ts += ssum[j * 16 + c]; tq += ssq[j * 16 + c]; }
        atomicAdd(&stats[c], ts);
        atomicAdd(&stats[16 + c], tq);
    }
}

// stats[0:16]=sum, [16:32]=sumsq -> [32:48]=scale, [48:64]=shift
__global__ void finalize_bn(float* __restrict__ stats,
                            const float* __restrict__ gamma,
                            const float* __restrict__ beta)
{
    int c = threadIdx.x;
    if (c < NCH) {
        const float invN = 1.0f / (float)NPTS;
        float mean = stats[c] * invN;
        float var  = stats[16 + c] * invN - mean * mean;
        float sc   = gamma[c] * rsqrtf(var + BN_EPS);
        stats[32 + c] = sc;
        stats[48 + c] = beta[c] - mean * sc;
    }
}

__device__ __forceinline__ float elu1(float v) {
    return v > 0.f ? v : (__expf(v) - 1.f);
}

// in-place BN+ELU over N*16 elements
__global__ void bn_elu_apply(float* __restrict__ x, const float* __restrict__ stats)
{
    long i = (long)blockIdx.x * blockDim.x + threadIdx.x;
    if (i < (long)NPTS * NCH) {
        int c = (int)(i & 15);
        x[i] = elu1(x[i] * stats[32 + c] + stats[48 + c]);
    }
}

// ---------------- conv2: WMMA f32 16x16x4, one (k, 16-row tile) per wave ----------------
__global__ void conv2_wmma(const float* __restrict__ h1,
                           const float* __restrict__ W2,
                           const int* __restrict__ in_map,
                           const int* __restrict__ out_map,
                           float* __restrict__ out2)
{
    const int lane   = threadIdx.x & 31;
    const int wib    = threadIdx.x >> 5;
    const long wid   = (long)blockIdx.x * (blockDim.x >> 5) + wib;
    const long total = (long)KVOL * TILES_PER_K;
    if (wid >= total) return;                 // whole-wave exit: EXEC stays all-1s for WMMA

    const int k    = (int)(wid / TILES_PER_K);
    const int base = (int)(wid % TILES_PER_K) * 16;

    const int* inm  = in_map  + (size_t)k * NPTS + base;
    const int* outm = out_map + (size_t)k * NPTS + base;
    const float* Bk = W2 + (size_t)k * 256;   // [16 Cin][16 Cout] row-major

    const int half = lane >> 4;               // 0: K pair {0,1}; 1: K pair {2,3}
    const int l16  = lane & 15;

    // A gather: lane l16 owns row M=l16 of the tile; half selects K sub-pair
    const int arow = inm[l16];
    const float* aptr = h1 + (size_t)arow * NCH + 2 * half;

    v8f acc = {0.f, 0.f, 0.f, 0.f, 0.f, 0.f, 0.f, 0.f};
#pragma unroll
    for (int q = 0; q < 4; ++q) {             // K = 4q .. 4q+3
        v2f a, b;
        a.x = aptr[4 * q + 0];
        a.y = aptr[4 * q + 1];
        const int kb = 4 * q + 2 * half;      // B rows striped: half-wave holds K, K+1
        b.x = Bk[(kb + 0) * NCH + l16];
        b.y = Bk[(kb + 1) * NCH + l16];
        // D = A(16x4) x B(4x16) + C ; chained f32 accumulation, K=16 total
        acc = __builtin_amdgcn_wmma_f32_16x16x4_f32(
                  false, a, false, b, (short)0, acc, false, false);
    }

    // C/D layout: VGPR j -> row (j + 8*half), col l16
#pragma unroll
    for (int j = 0; j < 8; ++j) {
        const int m    = j + 8 * half;
        const long orow = (long)outm[m] * NCH;
        atomicAdd(&out2[orow + l16], acc[j]);
    }
}

// ---------------- BN+ELU + 1x1 cls conv + pruning + concat ----------------
__global__ void cls_prune(const float* __restrict__ buf2,
                          const float* __restrict__ stats,
                          const float* __restrict__ Wcls,
                          const float* __restrict__ bcls,
                          float* __restrict__ out)
{
    int r = blockIdx.x * blockDim.x + threadIdx.x;
    if (r >= NPTS) return;
    float h[NCH];
    float cls = bcls[0];
#pragma unroll
    for (int c = 0; c < NCH; ++c) {
        float v = buf2[(long)r * NCH + c] * stats[32 + c] + stats[48 + c];
        v = elu1(v);
        h[c] = v;
        cls += v * Wcls[c];
    }
    const bool keep = cls > 0.f;
    float* o = out + (long)r * (NCH + 1);
#pragma unroll
    for (int c = 0; c < NCH; ++c) o[c] = keep ? h[c] : 0.f;
    o[NCH] = cls;
}

extern "C" void kernel_launch(void* const* d_in, const int* in_sizes, int n_in,
                              void* d_out, int out_size, void* d_ws, size_t ws_size,
                              hipStream_t stream)
{
    const float* feats  = (const float*)d_in[0];
    const float* W1     = (const float*)d_in[1];
    const float* gamma1 = (const float*)d_in[2];
    const float* beta1  = (const float*)d_in[3];
    const float* W2     = (const float*)d_in[4];
    const float* gamma2 = (const float*)d_in[5];
    const float* beta2  = (const float*)d_in[6];
    const float* Wcls   = (const float*)d_in[7];
    const float* bcls   = (const float*)d_in[8];
    const int* in_map1  = (const int*)d_in[9];
    const int* out_map1 = (const int*)d_in[10];
    const int* in_map2  = (const int*)d_in[11];
    const int* out_map2 = (const int*)d_in[12];
    float* out = (float*)d_out;

    const size_t featBytes = (size_t)NPTS * NCH * sizeof(float);   // 64 MB
    float* buf1   = (float*)d_ws;                                  // conv1 out -> h1 in place
    float* buf2   = (float*)((char*)d_ws + featBytes);             // conv2 out
    float* statsA = (float*)((char*)d_ws + 2 * featBytes);         // 64 floats
    float* statsB = statsA + 64;                                   // 64 floats

    // zero accumulators + stats (graph-capture safe)
    hipMemsetAsync(d_ws, 0, 2 * featBytes + 128 * sizeof(float), stream);

    // stage 1: conv (1 -> 16)
    {
        const long total = (long)KVOL * NPTS;
        int blocks = (int)((total + 255) / 256);
        conv1_scatter<<<blocks, 256, 0, stream>>>(feats, W1, in_map1, out_map1, buf1, total);
    }
    reduce_stats<<<2048, 256, 0, stream>>>(buf1, statsA);
    finalize_bn<<<1, 32, 0, stream>>>(statsA, gamma1, beta1);
    {
        long n = (long)NPTS * NCH;
        bn_elu_apply<<<(int)((n + 255) / 256), 256, 0, stream>>>(buf1, statsA);
    }

    // stage 2: conv (16 -> 16) via V_WMMA_F32_16X16X4_F32
    {
        const long waves  = (long)KVOL * TILES_PER_K;   // 1,687,500
        const int wavesPerBlock = 8;                    // 256 threads
        int blocks = (int)((waves + wavesPerBlock - 1) / wavesPerBlock);
        conv2_wmma<<<blocks, 256, 0, stream>>>(buf1, W2, in_map2, out_map2, buf2);
    }
    reduce_stats<<<2048, 256, 0, stream>>>(buf2, statsB);
    finalize_bn<<<1, 32, 0, stream>>>(statsB, gamma2, beta2);

    // BN+ELU + cls + prune, fused into the output pass
    cls_prune<<<(NPTS + 255) / 256, 256, 0, stream>>>(buf2, statsB, Wcls, bcls, out);
}